// LaplacianPyramid_83751862272202
// MI455X (gfx1250) — compile-verified
//
#include <hip/hip_runtime.h>
#include <math.h>

// Problem constants (reference: x [2, 64, 64, 64, 64] fp32)
#define CCH   64
#define NVOL  (64*64*64)          // 262144 spatial elements per (b,c)
#define NBAT  2
#define NROWS (NBAT*CCH)          // 128
#define SFULL (NBAT*CCH*NVOL)     // 33554432 floats per full tensor

#define KB     64                 // K-block staged in LDS per iteration
#define LPAD   68                 // LDS row stride (avoid bank conflicts)
#define NCHUNK 64                 // K-chunks per (b,level)
#define KC     (NVOL/NCHUNK)      // 4096

typedef __attribute__((ext_vector_type(2))) float v2f;
typedef __attribute__((ext_vector_type(8))) float v8f;

struct Taps { float w[7]; int r; };
struct Ptrs { const float* a[3]; const float* b[3]; };

// ---------------------------------------------------------------------------
// Separable 1D Gaussian blur along one axis of the 64^3 volume.
// shift: 0 -> innermost (d, stride 1), 6 -> w (stride 64), 12 -> h (stride 4096)
// Zero ('constant') boundary handling to match SAME zero padding.
// ---------------------------------------------------------------------------
__global__ void blur1d_kernel(const float* __restrict__ src,
                              float* __restrict__ dst,
                              int shift, Taps taps) {
    int i = blockIdx.x * blockDim.x + threadIdx.x;   // grid == SFULL/256 exactly
    int c = (i >> shift) & 63;
    int stride = 1 << shift;
    float acc = 0.f;
    for (int t = -taps.r; t <= taps.r; ++t) {
        int cc = c + t;
        if (cc >= 0 && cc < 64)
            acc += taps.w[t + taps.r] * src[i + t * stride];
    }
    dst[i] = acc;
}

// ---------------------------------------------------------------------------
// Online softmax statistics per (b,c) row of length NVOL.
// L = A - B (B may be null). Writes st[row*2] = rowmax, st[row*2+1] = 1/sumexp.
// ---------------------------------------------------------------------------
__global__ void rowstats_kernel(const float* __restrict__ A,
                                const float* __restrict__ Bm,
                                float* __restrict__ st) {
    int row = blockIdx.x;
    int tid = threadIdx.x;
    size_t base = (size_t)row * NVOL;
    float m = -3.402823466e38f, s = 0.f;
    for (int n = tid; n < NVOL; n += 256) {
        float v = A[base + n];
        if (Bm) v -= Bm[base + n];
        if (v > m) { s *= __expf(m - v); m = v; }
        s += __expf(v - m);
    }
    __shared__ float sm[256], ss[256];
    sm[tid] = m; ss[tid] = s;
    __syncthreads();
    for (int off = 128; off > 0; off >>= 1) {
        if (tid < off) {
            float m2 = sm[tid + off], s2 = ss[tid + off];
            float M = fmaxf(sm[tid], m2);
            ss[tid] = ss[tid] * __expf(sm[tid] - M) + s2 * __expf(m2 - M);
            sm[tid] = M;
        }
        __syncthreads();
    }
    if (tid == 0) { st[row * 2] = sm[0]; st[row * 2 + 1] = 1.0f / ss[0]; }
}

__global__ void zero_kernel(float* __restrict__ p, int n) {
    int i = blockIdx.x * blockDim.x + threadIdx.x;
    if (i < n) p[i] = 0.f;
}

// ---------------------------------------------------------------------------
// Channel attention GEMM: Acc[l][b] += P(64xKC) * L^T(KCx64), fp32 WMMA.
// One workgroup = one (level, batch, K-chunk). 8 waves x 2 tiles = 16 tiles.
// L block staged once in LDS; P block = exp(L - m) * invZ staged alongside.
// ---------------------------------------------------------------------------
__global__ void attn_gemm_kernel(Ptrs p,
                                 const float* __restrict__ stats,
                                 float* __restrict__ accOut) {
    __shared__ float Llds[64 * LPAD];
    __shared__ float Plds[64 * LPAD];

    int comb  = blockIdx.x % 6;          // (level, batch)
    int chunk = blockIdx.x / 6;
    int ell = comb >> 1;
    int b   = comb & 1;

    const float* A  = p.a[ell];
    const float* Bm = p.b[ell];
    const float* st = stats + ell * 256 + b * 128;   // [c][2] = {m, invZ}

    size_t gbase = (size_t)b * CCH * NVOL;
    int tid  = threadIdx.x;
    int lane = tid & 31;
    int wave = tid >> 5;
    int hl = lane >> 4;        // which half of the wave (K split)
    int r  = lane & 15;        // row-within-tile / col-within-tile

    int t0 = wave, t1 = wave + 8;
    int m0 = (t0 >> 2) * 16, n0 = (t0 & 3) * 16;
    int m1 = (t1 >> 2) * 16, n1 = (t1 & 3) * 16;

    v8f acc0 = {}; v8f acc1 = {};

    int k0 = chunk * KC;
    for (int kb = k0; kb < k0 + KC; kb += KB) {
        // Stage 64 x KB block of L (and softmaxed P) into LDS, coalesced on n.
        for (int i = tid; i < 64 * KB; i += 256) {
            int row = i >> 6;
            int col = i & 63;
            size_t g = gbase + (size_t)row * NVOL + kb + col;
            float v = A[g];
            if (Bm) v -= Bm[g];
            Llds[row * LPAD + col] = v;
            Plds[row * LPAD + col] = __expf(v - st[row * 2]) * st[row * 2 + 1];
        }
        __syncthreads();

        for (int k = 0; k < KB; k += 4) {
            int kk = k + 2 * hl;   // lanes 0-15 -> K=k,k+1; lanes 16-31 -> K=k+2,k+3
            v2f af0, bf0, af1, bf1;
            af0.x = Plds[(m0 + r) * LPAD + kk];
            af0.y = Plds[(m0 + r) * LPAD + kk + 1];
            bf0.x = Llds[(n0 + r) * LPAD + kk];
            bf0.y = Llds[(n0 + r) * LPAD + kk + 1];
            acc0 = __builtin_amdgcn_wmma_f32_16x16x4_f32(
                false, af0, false, bf0, (short)0, acc0, false, false);

            af1.x = Plds[(m1 + r) * LPAD + kk];
            af1.y = Plds[(m1 + r) * LPAD + kk + 1];
            bf1.x = Llds[(n1 + r) * LPAD + kk];
            bf1.y = Llds[(n1 + r) * LPAD + kk + 1];
            acc1 = __builtin_amdgcn_wmma_f32_16x16x4_f32(
                false, af1, false, bf1, (short)0, acc1, false, false);
        }
        __syncthreads();
    }

    // D layout: vgpr j -> M = j + 8*hl (lanes>=16), N = n + r
    float* dst = accOut + (ell * 2 + b) * 64 * 64;
    #pragma unroll
    for (int j = 0; j < 8; ++j) {
        atomicAdd(&dst[(m0 + j + 8 * hl) * 64 + (n0 + r)], acc0[j]);
        atomicAdd(&dst[(m1 + j + 8 * hl) * 64 + (n1 + r)], acc1[j]);
    }
}

// ---------------------------------------------------------------------------
// out[b][c][:] = softmax_row(A0[b][c][:]) + A1[b][c][:] + A2[b][c][:]
// ---------------------------------------------------------------------------
__global__ void finalize_kernel(const float* __restrict__ acc,
                                float* __restrict__ out) {
    int b = blockIdx.x, c = threadIdx.x;
    const float* A0 = acc + (0 * 2 + b) * 4096 + c * 64;
    const float* A1 = acc + (1 * 2 + b) * 4096 + c * 64;
    const float* A2 = acc + (2 * 2 + b) * 4096 + c * 64;
    float m = -3.402823466e38f;
    for (int j = 0; j < 64; ++j) m = fmaxf(m, A0[j]);
    float s = 0.f;
    for (int j = 0; j < 64; ++j) s += __expf(A0[j] - m);
    float inv = 1.f / s;
    float* o = out + (b * 64 + c) * 64;
    for (int j = 0; j < 64; ++j)
        o[j] = __expf(A0[j] - m) * inv + A1[j] + A2[j];
}

// ---------------------------------------------------------------------------
static void make_taps(int ks, double sigma, Taps* t) {
    int r = ks / 2;
    t->r = r;
    double w[7], sum = 0.0;
    for (int i = 0; i < ks; ++i) {
        double a = (double)(i - r);
        w[i] = exp(-0.5 * (a / sigma) * (a / sigma));
        sum += w[i];
    }
    for (int i = 0; i < 7; ++i)
        t->w[i] = (i < ks) ? (float)(w[i] / sum) : 0.f;
}

extern "C" void kernel_launch(void* const* d_in, const int* in_sizes, int n_in,
                              void* d_out, int out_size, void* d_ws, size_t ws_size,
                              hipStream_t stream) {
    const float* x = (const float*)d_in[0];
    float* out = (float*)d_out;

    // Workspace layout (floats): T1 (=G1), T2 (=G2), T3 (scratch), stats, acc
    float* T1    = (float*)d_ws;
    float* T2    = T1 + SFULL;
    float* T3    = T2 + SFULL;
    float* stats = T3 + SFULL;        // 3*128*2 = 768 floats used
    float* acc   = stats + 1024;      // 3*2*64*64 = 24576 floats

    Taps t1, t2;
    make_taps(3, 1.6, &t1);
    make_taps(5, 1.6 * pow(2.0, 1.0 / 3.0), &t2);

    dim3 blk(256);
    dim3 gBlur(SFULL / 256);

    // G1 = blur3(x)   (separable: d, w, h passes)
    blur1d_kernel<<<gBlur, blk, 0, stream>>>(x,  T2, 0,  t1);
    blur1d_kernel<<<gBlur, blk, 0, stream>>>(T2, T3, 6,  t1);
    blur1d_kernel<<<gBlur, blk, 0, stream>>>(T3, T1, 12, t1);
    // G2 = blur5(G1)
    blur1d_kernel<<<gBlur, blk, 0, stream>>>(T1, T2, 0,  t2);
    blur1d_kernel<<<gBlur, blk, 0, stream>>>(T2, T3, 6,  t2);
    blur1d_kernel<<<gBlur, blk, 0, stream>>>(T3, T2, 12, t2);
    // (ks=7 level of the pyramid is never consumed by the reference output)

    // Softmax row statistics for L0 = x, L1 = x - G1, L2 = G1 - G2
    rowstats_kernel<<<dim3(NROWS), blk, 0, stream>>>(x,  nullptr, stats + 0);
    rowstats_kernel<<<dim3(NROWS), blk, 0, stream>>>(x,  T1,      stats + 256);
    rowstats_kernel<<<dim3(NROWS), blk, 0, stream>>>(T1, T2,      stats + 512);

    zero_kernel<<<dim3(96), blk, 0, stream>>>(acc, 3 * 2 * 64 * 64);

    Ptrs p;
    p.a[0] = x;  p.b[0] = nullptr;
    p.a[1] = x;  p.b[1] = T1;
    p.a[2] = T1; p.b[2] = T2;
    attn_gemm_kernel<<<dim3(6 * NCHUNK), blk, 0, stream>>>(p, stats, acc);

    finalize_kernel<<<dim3(2), dim3(64), 0, stream>>>(acc, out);
}